// NextStateHyperbolicLoss_38697655337257
// MI455X (gfx1250) — compile-verified
//
#include <hip/hip_runtime.h>

typedef __attribute__((ext_vector_type(2))) float v2f;
typedef __attribute__((ext_vector_type(8))) float v8f;

#define MARGIN_F  0.2f
#define NEG_W_F   0.5f
#define EPS_DEN   1e-15f
#define X2CLAMP   0.999999f   /* (1 - 1e-6) / C, C = 1 */

__device__ __forceinline__ float poincare_acosh(float num, float den) {
    // z = 1 + max(num/max(den,EPS), 1+1e-7);  arccosh(z) = log(z + sqrt(z^2-1))
    float r = num / fmaxf(den, EPS_DEN);
    float z = 1.0f + fmaxf(r, 1.0f + 1e-7f);
    return logf(z + sqrtf(z * z - 1.0f));
}

// ---- dynamic-LDS layout (main kernel uses NO static LDS => dynamic base offset 0)
#define OFF_X    0u        // 64 floats   (256 B)
#define OFF_T    256u      // 8*32 floats (1024 B)
#define OFF_RED  1280u     // 256 floats  (1024 B)
#define OFF_Y0   2304u     // 128*72 floats (36864 B)
#define OFF_Y1   39168u    // 128*72 floats (36864 B)
#define LDS_TOTAL 76032u

template <int DH, int LSEQ, int CHUNK>
__global__ __launch_bounds__(256) void hyper_loss_main(
    const float* __restrict__ h_curr,
    const float* __restrict__ demo_h,
    const int*   __restrict__ align_idx,
    const int*   __restrict__ lookahead,
    float* __restrict__ ws,
    int Bdim, int Ndim, int Adim)
{
    constexpr int STRIDE = 72;   // padded LDS row stride in floats (288 B, float4-aligned)
    constexpr int NC     = LSEQ / CHUNK;

    extern __shared__ __align__(16) char smem[];
    float* sX   = (float*)(smem + OFF_X);
    float* sT   = (float*)(smem + OFF_T);    // per-wave 32 floats: [0..15]=y2, [16..31]=dot
    float* sRed = (float*)(smem + OFF_RED);

    const int samp = blockIdx.x;
    const int b    = samp / Adim;
    const int n0   = align_idx[2 * samp + 0];
    const int l0   = align_idx[2 * samp + 1];
    const int lt   = l0 + lookahead[0];
    const bool valid = (lt >= 0) && (lt < LSEQ);
    const int ltc  = lt < 0 ? 0 : (lt > LSEQ - 1 ? LSEQ - 1 : lt);

    const int tid  = threadIdx.x;
    const int wave = tid >> 5;
    const int lane = tid & 31;

    const float* __restrict__ seq = demo_h + (size_t)(b * Ndim + n0) * (size_t)LSEQ * DH;
    const float* __restrict__ tgt = seq + (size_t)ltc * DH;
    const unsigned long long seqAddr = (unsigned long long)(uintptr_t)seq;

    // ---- async stage: each thread issues 8 x B128 (128 rows x 64 f32 per chunk)
    auto issue_chunk = [&](int c, unsigned ldsBase) {
#pragma unroll
        for (int it = 0; it < (CHUNK * DH / 4) / 256; ++it) {
            int e4 = tid + it * 256;
            int r  = e4 >> 4;     // DH/4 = 16 float4 per row
            int c4 = e4 & 15;
            unsigned gOff = (unsigned)(((c * CHUNK + r) * DH + c4 * 4) * 4);   // bytes
            unsigned lOff = ldsBase + (unsigned)(r * (STRIDE * 4) + c4 * 16);  // bytes
            asm volatile("global_load_async_to_lds_b128 %0, %1, %2"
                         :
                         : "v"(lOff), "v"(gOff), "s"(seqAddr)
                         : "memory");
        }
    };

    // prologue: chunk 0 in flight while we do d_pos
    issue_chunk(0, OFF_Y0);

    if (tid < DH) sX[tid] = h_curr[b * DH + tid];
    __syncthreads();

    // d_pos computed redundantly per thread (64 cached loads; negligible)
    float x2r = 0.f, t2r = 0.f, dp2 = 0.f;
#pragma unroll
    for (int k = 0; k < DH; ++k) {
        float xv = sX[k], tv = tgt[k];
        x2r += xv * xv;
        t2r += tv * tv;
        float df = xv - tv;
        dp2 += df * df;
    }
    const float den_p = (1.f - fminf(x2r, X2CLAMP)) * (1.f - fminf(t2r, X2CLAMP));
    const float d_pos = poincare_acosh(2.f * dp2, den_p);

    float hingeSum = 0.f;
    const bool hi      = (lane >= 16);
    const int  myrow   = lane & 15;
    const int  tileRow = wave * 16;
    const float4* sX4  = (const float4*)sX;

    for (int chunk = 0; chunk < NC; ++chunk) {
        // my async loads for this chunk are done; barrier makes everyone's visible
        asm volatile("s_wait_asynccnt 0x0" ::: "memory");
        __syncthreads();

        // overlap: stream next chunk into the other buffer while we compute
        if (chunk + 1 < NC)
            issue_chunk(chunk + 1, ((chunk + 1) & 1) ? OFF_Y1 : OFF_Y0);

        const float* sY = (const float*)(smem + ((chunk & 1) ? OFF_Y1 : OFF_Y0));

        // WMMA chains: cY = Y * Y^T (need diag), cX = Y * [x x ... x] (need one column)
        v8f cY = {0.f, 0.f, 0.f, 0.f, 0.f, 0.f, 0.f, 0.f};
        v8f cX = {0.f, 0.f, 0.f, 0.f, 0.f, 0.f, 0.f, 0.f};
#pragma unroll
        for (int ks = 0; ks < DH / 4; ++ks) {
            float4 yv = *(const float4*)(&sY[(tileRow + myrow) * STRIDE + ks * 4]);
            float4 xv = sX4[ks];  // uniform broadcast
            // A (16x4 f32): lanes 0-15 hold K={0,1}; lanes 16-31 hold K={2,3}
            v2f a  = { hi ? yv.z : yv.x, hi ? yv.w : yv.y };
            // B (4x16 f32): v0: lanes0-15=K0 row, lanes16-31=K1; v1: K2 / K3
            v2f by = { hi ? yv.y : yv.x, hi ? yv.w : yv.z };
            v2f bx = { hi ? xv.y : xv.x, hi ? xv.w : xv.z };
            cY = __builtin_amdgcn_wmma_f32_16x16x4_f32(false, a, false, by, (short)0, cY, false, false);
            cX = __builtin_amdgcn_wmma_f32_16x16x4_f32(false, a, false, bx, (short)0, cX, false, false);
        }

        // Extract diag(cY) and one column of cX per the 16x16 f32 C/D layout:
        // lane L<16 has rows 0-7 (VGPR r -> M=r, N=L); lane L>=16 has rows 8-15.
#pragma unroll
        for (int r = 0; r < 8; ++r) {
            if (lane == r)      sT[wave * 32 + r]      = cY[r];   // diag M=N=r
            if (lane == 24 + r) sT[wave * 32 + 8 + r]  = cY[r];   // diag M=N=8+r
            if (lane == 0)      sT[wave * 32 + 16 + r] = cX[r];   // dot rows 0-7
            if (lane == 16)     sT[wave * 32 + 24 + r] = cX[r];   // dot rows 8-15
        }
        __syncthreads();

        if (lane < 16) {
            float y2r  = sT[wave * 32 + lane];
            float dotv = sT[wave * 32 + 16 + lane];
            int   j    = chunk * CHUNK + tileRow + lane;
            float diff2 = x2r + y2r - 2.f * dotv;   // ||x-y||^2 = x2 + y2 - 2 x.y
            float den   = (1.f - fminf(x2r, X2CLAMP)) * (1.f - fminf(y2r, X2CLAMP));
            float dneg  = poincare_acosh(2.f * diff2, den);
            if (valid && j > lt)
                hingeSum += fmaxf(MARGIN_F - dneg + d_pos, 0.f);
        }
    }

    // Block tree reduction (deterministic order)
    __syncthreads();
    sRed[tid] = hingeSum;
    __syncthreads();
    for (int s = 128; s > 0; s >>= 1) {
        if (tid < s) sRed[tid] += sRed[tid + s];
        __syncthreads();
    }
    if (tid == 0) {
        float vf  = valid ? 1.f : 0.f;
        float cnt = valid ? (float)(LSEQ - 1 - lt) : 0.f;  // #{ j : j > lt }
        float has = cnt > 0.f ? 1.f : 0.f;
        float per = sRed[0] / fmaxf(cnt, 1.f);
        int S = gridDim.x;
        ws[samp]         = d_pos * vf;
        ws[S + samp]     = vf;
        ws[2 * S + samp] = per * has;
        ws[3 * S + samp] = has;
    }
}

__global__ __launch_bounds__(256) void hyper_loss_finalize(
    const float* __restrict__ ws, int S, float* __restrict__ out)
{
    __shared__ float4 sR[256];
    int tid = threadIdx.x;
    float a = 0.f, b = 0.f, c = 0.f, d = 0.f;
    for (int i = tid; i < S; i += 256) {
        a += ws[i];
        b += ws[S + i];
        c += ws[2 * S + i];
        d += ws[3 * S + i];
    }
    sR[tid] = make_float4(a, b, c, d);
    __syncthreads();
    for (int s = 128; s > 0; s >>= 1) {
        if (tid < s) {
            float4 o = sR[tid + s];
            float4 m = sR[tid];
            m.x += o.x; m.y += o.y; m.z += o.z; m.w += o.w;
            sR[tid] = m;
        }
        __syncthreads();
    }
    if (tid == 0) {
        float pos = sR[0].x / fmaxf(sR[0].y, 1.f);
        float neg = sR[0].z / fmaxf(sR[0].w, 1.f);
        out[0] = pos + NEG_W_F * neg;
    }
}

extern "C" void kernel_launch(void* const* d_in, const int* in_sizes, int n_in,
                              void* d_out, int out_size, void* d_ws, size_t ws_size,
                              hipStream_t stream) {
    const float* h_curr    = (const float*)d_in[0];
    const float* demo_h    = (const float*)d_in[1];
    const int*   align_idx = (const int*)d_in[2];
    const int*   lookahead = (const int*)d_in[3];

    constexpr int DH = 64, LSEQ = 1024, CHUNK = 128;
    const int B  = in_sizes[0] / DH;
    const int NL = in_sizes[1] / (B * DH);
    const int N  = NL / LSEQ;
    const int A  = in_sizes[2] / (B * 2);
    const int S  = B * A;

    float* ws = (float*)d_ws;  // needs 4*S floats (8 KB) — all entries written each call

    hipLaunchKernelGGL((hyper_loss_main<DH, LSEQ, CHUNK>),
                       dim3(S), dim3(256), LDS_TOTAL, stream,
                       h_curr, demo_h, align_idx, lookahead, ws, B, N, A);
    hipLaunchKernelGGL(hyper_loss_finalize,
                       dim3(1), dim3(256), 0, stream,
                       ws, S, (float*)d_out);
}